// value_wise_projector_56324201120039
// MI455X (gfx1250) — compile-verified
//
#include <hip/hip_runtime.h>

// ---------------- problem constants ----------------
#define NBINS 256
#define EPS_F 1e-5f
#define SLICE (96 * 96 * 96)          // 884736 elements per (n,c) slice
#define NC_TOTAL 32                   // N*C = 4*8
#define C_DIM 8
#define TPB 256                       // 8 waves (wave32)

#define TILE 4096                     // floats per TDM tile (16 KB)
#define TILES_PER_SLICE (SLICE / TILE)        // 216
#define NBLOCKS_STATS (NC_TOTAL * TILES_PER_SLICE)  // 6912

#define TILES_PER_BLK 4
#define BLKS_PER_SLICE (TILES_PER_SLICE / TILES_PER_BLK)  // 54

typedef float v4f __attribute__((ext_vector_type(4)));
typedef float v2f __attribute__((ext_vector_type(2)));
typedef unsigned int u32x4 __attribute__((ext_vector_type(4)));
typedef int i32x4 __attribute__((ext_vector_type(4)));
typedef int i32x8 __attribute__((ext_vector_type(8)));

#if defined(__gfx1250__) && __has_builtin(__builtin_amdgcn_tensor_load_to_lds)
#define HAVE_TDM 1
#else
#define HAVE_TDM 0
#endif

__device__ __forceinline__ void wait_tensorcnt0() {
#if defined(__gfx1250__)
#if __has_builtin(__builtin_amdgcn_s_wait_tensorcnt)
  __builtin_amdgcn_s_wait_tensorcnt(0);
#else
  asm volatile("s_wait_tensorcnt 0x0" ::: "memory");
#endif
#endif
}

#if HAVE_TDM
// Issue a 1-D TDM copy of TILE floats: global -> LDS. One call per wave
// (EXEC is ignored by TENSOR_LOAD_TO_LDS; we gate so only wave 0 issues).
__device__ __forceinline__ void tdm_load_tile(const float* gsrc, float* lds_dst) {
  unsigned long long ga = (unsigned long long)gsrc;
  unsigned lv = (unsigned)(unsigned long long)lds_dst;  // low 32 bits = LDS byte addr

  u32x4 g0;
  g0.x = 1u;                                   // count=1, user descriptor
  g0.y = lv;                                   // lds_addr
  g0.z = (unsigned)ga;                         // global_addr[31:0]
  g0.w = (unsigned)((ga >> 32) & 0x01FFFFFFull) | (2u << 30);  // addr[56:32] | type=2

  i32x8 g1;
  g1.s0 = (int)(2u << 16);                     // workgroup_mask=0, data_size=2 (4B), no flags
  g1.s1 = (int)(((unsigned)TILE & 0xFFFFu) << 16);   // tensor_dim0[15:0] @bits63:48
  g1.s2 = (int)((((unsigned)TILE >> 16) & 0xFFFFu)   // tensor_dim0[31:16] @bits79:64
               | (1u << 16));                        // tensor_dim1[15:0] = 1
  g1.s3 = (int)(((unsigned)TILE & 0xFFFFu) << 16);   // tensor_dim1 hi=0 | tile_dim0=TILE
  g1.s4 = 0;                                   // tile_dim1=0 (unused), tile_dim2=0
  g1.s5 = (int)TILE;                           // tensor_dim0_stride[31:0]
  g1.s6 = 0;                                   // stride0 hi | dim1_stride lo
  g1.s7 = 0;                                   // dim1_stride hi

  i32x4 z4;
  z4.x = 0; z4.y = 0; z4.z = 0; z4.w = 0;      // groups 2/3 unused (<=2D tensor)
  i32x8 z8;
  z8.s0 = 0; z8.s1 = 0; z8.s2 = 0; z8.s3 = 0;
  z8.s4 = 0; z8.s5 = 0; z8.s6 = 0; z8.s7 = 0;

  // 6-arg toolchain form: (g0, g1, g2, g3, g4, cpol)
  __builtin_amdgcn_tensor_load_to_lds(g0, g1, z4, z4, z8, 0);
}
#endif

// ---------------- element op: normalize + 256-bin piecewise-linear lookup ---
__device__ __forceinline__ float proj1(float xx, float s255, float t255,
                                       const float* __restrict__ pmap) {
  const float scaled = fmaf(xx, s255, t255);     // normed * (NBINS-1)
  const float lo = floorf(scaled);
  const float w = scaled - lo;
  int li = (int)lo;
  int ui = (int)ceilf(scaled);
  li = min(max(li, 0), NBINS - 1);
  ui = min(max(ui, 0), NBINS - 1);
  const float lv = pmap[li];
  const float uv = pmap[ui];
  return fmaf(uv - lv, w, lv);                   // lv*(1-w)+uv*w
}

// ---------------- pass 2: TDM-staged apply ----------------
// 2-D grid: blockIdx.x = block-in-slice (0..53), blockIdx.y = nc (0..31).
// Avoids the div-by-54 magic-number prologue entirely.
__global__ __launch_bounds__(TPB) void apply_kernel(
    const float* __restrict__ x, const float* __restrict__ gamma,
    const float* __restrict__ beta, const float* __restrict__ pmap_g,
    const v2f* __restrict__ stats, float* __restrict__ out) {
  __shared__ __align__(16) float pmap[NBINS];
  __shared__ __align__(16) float buf[2][TILE];

  const int nc = blockIdx.y;
  const size_t base =
      (size_t)nc * SLICE + (size_t)blockIdx.x * (TILE * TILES_PER_BLK);
  const int wid = threadIdx.x >> 5;

#if HAVE_TDM
  if (wid == 0) tdm_load_tile(x + base, &buf[0][0]);  // DMA in flight ASAP
#endif

  pmap[threadIdx.x] = pmap_g[threadIdx.x];       // TPB == NBINS

  const int c = nc & (C_DIM - 1);
  const v2f st = stats[nc];
  const float scale = st.y * gamma[c];           // rstd * gamma
  const float shift = fmaf(-st.x, scale, beta[c]);
  const float s255 = scale * (float)(NBINS - 1); // fold *(NBINS-1) into the FMA
  const float t255 = shift * (float)(NBINS - 1);

#if HAVE_TDM
  for (int t = 0; t < TILES_PER_BLK; ++t) {
    const int cur = t & 1;
    if (wid == 0) wait_tensorcnt0();             // tile t resident in LDS
    __syncthreads();
    if (wid == 0 && (t + 1) < TILES_PER_BLK)     // overlap DMA of t+1 with compute
      tdm_load_tile(x + base + (size_t)(t + 1) * TILE, &buf[1 - cur][0]);

    const v4f* __restrict__ bv = (const v4f*)&buf[cur][0];
    v4f* __restrict__ ov = (v4f*)(out + base + (size_t)t * TILE);
#pragma unroll
    for (int i = 0; i < TILE / (TPB * 4); ++i) {
      const v4f v = bv[threadIdx.x + i * TPB];
      v4f r;
      r.x = proj1(v.x, s255, t255, pmap);
      r.y = proj1(v.y, s255, t255, pmap);
      r.z = proj1(v.z, s255, t255, pmap);
      r.w = proj1(v.w, s255, t255, pmap);
      __builtin_nontemporal_store(r, &ov[threadIdx.x + i * TPB]);
    }
    __syncthreads();                             // buf[cur] reusable for tile t+2
  }
#else
  // Fallback: cooperative staging with plain b128 loads.
  for (int t = 0; t < TILES_PER_BLK; ++t) {
    const int cur = t & 1;
    const v4f* __restrict__ gv = (const v4f*)(x + base + (size_t)t * TILE);
    v4f* __restrict__ lv = (v4f*)&buf[cur][0];
#pragma unroll
    for (int i = 0; i < TILE / (TPB * 4); ++i)
      lv[threadIdx.x + i * TPB] = gv[threadIdx.x + i * TPB];
    __syncthreads();
    const v4f* __restrict__ bv = (const v4f*)&buf[cur][0];
    v4f* __restrict__ ov = (v4f*)(out + base + (size_t)t * TILE);
#pragma unroll
    for (int i = 0; i < TILE / (TPB * 4); ++i) {
      const v4f v = bv[threadIdx.x + i * TPB];
      v4f r;
      r.x = proj1(v.x, s255, t255, pmap);
      r.y = proj1(v.y, s255, t255, pmap);
      r.z = proj1(v.z, s255, t255, pmap);
      r.w = proj1(v.w, s255, t255, pmap);
      __builtin_nontemporal_store(r, &ov[threadIdx.x + i * TPB]);
    }
    __syncthreads();
  }
#endif
}

// ---------------- pass 1a: per-tile partial sums ----------------
__global__ __launch_bounds__(TPB) void stats_partial_kernel(
    const float* __restrict__ x, v2f* __restrict__ partials) {
  const int b = blockIdx.x;                       // tile id over whole array
  const v4f* __restrict__ pv = (const v4f*)(x + (size_t)b * TILE);

  float s = 0.0f, q = 0.0f;
#pragma unroll
  for (int i = 0; i < TILE / (TPB * 4); ++i) {    // 4 x b128 loads / thread
    v4f v = pv[threadIdx.x + i * TPB];
    s += v.x + v.y + v.z + v.w;
    q = fmaf(v.x, v.x, q); q = fmaf(v.y, v.y, q);
    q = fmaf(v.z, v.z, q); q = fmaf(v.w, v.w, q);
  }
  // wave32 reduction
#pragma unroll
  for (int off = 16; off > 0; off >>= 1) {
    s += __shfl_down(s, off, 32);
    q += __shfl_down(q, off, 32);
  }
  __shared__ v2f wsum[TPB / 32];
  const int wid = threadIdx.x >> 5, lane = threadIdx.x & 31;
  if (lane == 0) { v2f t; t.x = s; t.y = q; wsum[wid] = t; }
  __syncthreads();
  if (threadIdx.x == 0) {
    float S = 0.0f, Q = 0.0f;
#pragma unroll
    for (int w = 0; w < TPB / 32; ++w) { S += wsum[w].x; Q += wsum[w].y; }
    v2f t; t.x = S; t.y = Q;
    partials[b] = t;
  }
}

// ---------------- pass 1b: finalize mean/rstd per slice ----------------
__global__ __launch_bounds__(64) void stats_final_kernel(
    const v2f* __restrict__ partials, v2f* __restrict__ stats) {
  const int nc = blockIdx.x;
  float s = 0.0f, q = 0.0f;
  for (int i = threadIdx.x; i < TILES_PER_SLICE; i += 64) {
    v2f p = partials[nc * TILES_PER_SLICE + i];
    s += p.x; q += p.y;
  }
#pragma unroll
  for (int off = 16; off > 0; off >>= 1) {
    s += __shfl_down(s, off, 32);
    q += __shfl_down(q, off, 32);
  }
  __shared__ v2f sh[2];
  const int wid = threadIdx.x >> 5, lane = threadIdx.x & 31;
  if (lane == 0) { v2f t; t.x = s; t.y = q; sh[wid] = t; }
  __syncthreads();
  if (threadIdx.x == 0) {
    const float S = sh[0].x + sh[1].x;
    const float Q = sh[0].y + sh[1].y;
    const float inv_n = 1.0f / (float)SLICE;
    const float mean = S * inv_n;
    const float var = fmaf(-mean, mean, Q * inv_n);
    v2f t; t.x = mean; t.y = __frsqrt_rn(var + EPS_F);
    stats[nc] = t;
  }
}

// ---------------- host-side launch ----------------
extern "C" void kernel_launch(void* const* d_in, const int* in_sizes, int n_in,
                              void* d_out, int out_size, void* d_ws, size_t ws_size,
                              hipStream_t stream) {
  const float* x     = (const float*)d_in[0];
  const float* gamma = (const float*)d_in[1];
  const float* beta  = (const float*)d_in[2];
  const float* pmap  = (const float*)d_in[3];
  float* out = (float*)d_out;

  // workspace layout: [0,256) stats (32 x v2f), [256, 256+6912*8) partials
  v2f* stats    = (v2f*)d_ws;
  v2f* partials = (v2f*)((char*)d_ws + 256);

  stats_partial_kernel<<<NBLOCKS_STATS, TPB, 0, stream>>>(x, partials);
  stats_final_kernel<<<NC_TOTAL, 64, 0, stream>>>(partials, stats);
  apply_kernel<<<dim3(BLKS_PER_SLICE, NC_TOTAL), TPB, 0, stream>>>(
      x, gamma, beta, pmap, stats, out);
}